// Loss_DC_31482110279712
// MI455X (gfx1250) — compile-verified
//
#include <hip/hip_runtime.h>
#include <hip/hip_bf16.h>

typedef __attribute__((ext_vector_type(2))) float v2f;
typedef __attribute__((ext_vector_type(8))) float v8f;

#define N   4096
#define DIM 2048
#define FEPS 1e-18f
#define ICH 32              // i-chunks for column-sum pass
#define IROWS (N / ICH)     // 128 rows per chunk
#define BBLOCKS 2048        // bracket reduction blocks

// ---------------- per-row sum of squares: sq[i] = sum_c F[i][c]^2 ----------------
__global__ void __launch_bounds__(256)
rowsq_kernel(const float* __restrict__ F, float* __restrict__ sq) {
  __shared__ float red[256];
  const int row = blockIdx.x;
  const float* r = F + (size_t)row * DIM;
  float s = 0.f;
  for (int c = threadIdx.x; c < DIM; c += 256) { float v = r[c]; s += v * v; }
  red[threadIdx.x] = s;
  __syncthreads();
  for (int off = 128; off > 0; off >>= 1) {
    if (threadIdx.x < off) red[threadIdx.x] += red[threadIdx.x + off];
    __syncthreads();
  }
  if (threadIdx.x == 0) sq[row] = red[0];
}

// ---------------- Gram + distance: a[i][j] = sqrt(max(sq_i+sq_j-2*<f_i,f_j>,0)+EPS) ----
// One wave computes a 64x64 tile as 4x4 WMMA 16x16 subtiles using fp32 WMMA
// (V_WMMA_F32_16X16X4_F32). A-frag (16x4 fp32): lanes 0-15 hold K={0,1} in 2 VGPRs,
// lanes 16-31 hold K={2,3} -> one b64 load per lane. B of F*F^T has B(k,n)=F[n][k],
// which gives the identical per-lane addressing pattern on F's rows.
__global__ void __launch_bounds__(256)
gram_dist_kernel(const float* __restrict__ F, const float* __restrict__ sq,
                 float* __restrict__ a_out) {
  const int lane = threadIdx.x & 31;
  const int wave = threadIdx.x >> 5;
  const int tile = blockIdx.x * 8 + wave;          // 4096 tiles total
  const int tpr  = N / 64;                         // 64 tiles per row
  const int m0 = (tile / tpr) * 64;
  const int n0 = (tile % tpr) * 64;
  const int half = lane >> 4;                      // 0 -> K{0,1}, 1 -> K{2,3}
  const int l16  = lane & 15;

  v8f acc[4][4] = {};

  const float* aptr[4];
  const float* bptr[4];
#pragma unroll
  for (int i = 0; i < 4; ++i) {
    aptr[i] = F + (size_t)(m0 + i * 16 + l16) * DIM + half * 2;
    bptr[i] = F + (size_t)(n0 + i * 16 + l16) * DIM + half * 2;
  }

  for (int k = 0; k < DIM; k += 4) {
    v2f afrag[4], bfrag[4];
#pragma unroll
    for (int i = 0; i < 4; ++i) {
      afrag[i] = *(const v2f*)(aptr[i] + k);
      bfrag[i] = *(const v2f*)(bptr[i] + k);
    }
#pragma unroll
    for (int i = 0; i < 4; ++i)
#pragma unroll
      for (int j = 0; j < 4; ++j)
        acc[i][j] = __builtin_amdgcn_wmma_f32_16x16x4_f32(
            /*neg_a=*/false, afrag[i], /*neg_b=*/false, bfrag[j],
            /*c_mod=*/(short)0, acc[i][j], /*reuse_a=*/false, /*reuse_b=*/false);
  }

  // D layout: VGPR r holds M=r (lanes 0-15, N=l16) and M=r+8 (lanes 16-31, N=l16)
#pragma unroll
  for (int i = 0; i < 4; ++i) {
#pragma unroll
    for (int j = 0; j < 4; ++j) {
      const int gj = n0 + j * 16 + l16;
      const float sqj = sq[gj];
#pragma unroll
      for (int r = 0; r < 8; ++r) {
        const int gi = m0 + i * 16 + half * 8 + r;
        float g  = acc[i][j][r];
        float d2 = sq[gi] + sqj - 2.0f * g;
        d2 = fmaxf(d2, 0.0f) + FEPS;
        a_out[(size_t)gi * N + gj] = sqrtf(d2);
      }
    }
  }
}

// ---------------- column sums: rowsum[j] = sum_i a[i][j] (two deterministic passes) ----
__global__ void __launch_bounds__(256)
colsum_part_kernel(const float* __restrict__ M, float* __restrict__ part) {
  const int j  = blockIdx.x * 256 + threadIdx.x;
  const int ic = blockIdx.y;
  const float* p = M + (size_t)(ic * IROWS) * N + j;
  float s = 0.f;
  for (int i = 0; i < IROWS; ++i) s += p[(size_t)i * N];
  part[(size_t)ic * N + j] = s;
}

__global__ void __launch_bounds__(256)
colsum_final_kernel(const float* __restrict__ part, float* __restrict__ rowsum) {
  const int j = blockIdx.x * 256 + threadIdx.x;
  double s = 0.0;
  for (int ic = 0; ic < ICH; ++ic) s += (double)part[(size_t)ic * N + j];
  rowsum[j] = (float)s;
}

// ---------------- totals: tots = { sum(rowsumA), sum(rowsumB) } ----------------
__global__ void __launch_bounds__(256)
tot_kernel(const float* __restrict__ rA, const float* __restrict__ rB,
           double* __restrict__ tots) {
  __shared__ double redA[256], redB[256];
  double sA = 0.0, sB = 0.0;
  for (int j = threadIdx.x; j < N; j += 256) { sA += rA[j]; sB += rB[j]; }
  redA[threadIdx.x] = sA; redB[threadIdx.x] = sB;
  __syncthreads();
  for (int off = 128; off > 0; off >>= 1) {
    if (threadIdx.x < off) {
      redA[threadIdx.x] += redA[threadIdx.x + off];
      redB[threadIdx.x] += redB[threadIdx.x + off];
    }
    __syncthreads();
  }
  if (threadIdx.x == 0) { tots[0] = redA[0]; tots[1] = redB[0]; }
}

// ---------------- brackets: sum(A*B), sum(A*A), sum(B*B) with literal centering ------
// Reference quirk: row used twice -> cent(i,j) = a[i,j] - 2*row[j]/(n-2) + tot, diag=0.
__global__ void __launch_bounds__(256)
bracket_kernel(const float* __restrict__ A, const float* __restrict__ B,
               const float* __restrict__ rA, const float* __restrict__ rB,
               const double* __restrict__ tots, double* __restrict__ part) {
  const float rdiv = 1.0f / (float)(N - 2);
  const double tscale = 1.0 / ((double)(N - 1) * (double)(N - 2));
  const float tA = (float)(tots[0] * tscale);
  const float tB = (float)(tots[1] * tscale);

  double ab = 0.0, aa = 0.0, bb = 0.0;
  const size_t total = (size_t)N * N;
  for (size_t idx = (size_t)blockIdx.x * 256 + threadIdx.x; idx < total;
       idx += (size_t)BBLOCKS * 256) {
    const int i = (int)(idx >> 12);
    const int j = (int)(idx & (N - 1));
    float ca, cb;
    if (i == j) { ca = 0.f; cb = 0.f; }
    else {
      ca = A[idx] - 2.0f * rA[j] * rdiv + tA;
      cb = B[idx] - 2.0f * rB[j] * rdiv + tB;
    }
    ab += (double)ca * (double)cb;
    aa += (double)ca * (double)ca;
    bb += (double)cb * (double)cb;
  }

  __shared__ double red[256 * 3];
  red[threadIdx.x]       = ab;
  red[256 + threadIdx.x] = aa;
  red[512 + threadIdx.x] = bb;
  __syncthreads();
  for (int off = 128; off > 0; off >>= 1) {
    if (threadIdx.x < off) {
      red[threadIdx.x]       += red[threadIdx.x + off];
      red[256 + threadIdx.x] += red[256 + threadIdx.x + off];
      red[512 + threadIdx.x] += red[512 + threadIdx.x + off];
    }
    __syncthreads();
  }
  if (threadIdx.x == 0) {
    part[(size_t)blockIdx.x * 3 + 0] = red[0];
    part[(size_t)blockIdx.x * 3 + 1] = red[256];
    part[(size_t)blockIdx.x * 3 + 2] = red[512];
  }
}

// ---------------- final: -g_xy / sqrt(g_xx*g_yy + EPS) ----------------
__global__ void __launch_bounds__(256)
final_kernel(const double* __restrict__ part, float* __restrict__ out) {
  __shared__ double red[256 * 3];
  double ab = 0.0, aa = 0.0, bb = 0.0;
  for (int b = threadIdx.x; b < BBLOCKS; b += 256) {
    ab += part[(size_t)b * 3 + 0];
    aa += part[(size_t)b * 3 + 1];
    bb += part[(size_t)b * 3 + 2];
  }
  red[threadIdx.x]       = ab;
  red[256 + threadIdx.x] = aa;
  red[512 + threadIdx.x] = bb;
  __syncthreads();
  for (int off = 128; off > 0; off >>= 1) {
    if (threadIdx.x < off) {
      red[threadIdx.x]       += red[threadIdx.x + off];
      red[256 + threadIdx.x] += red[256 + threadIdx.x + off];
      red[512 + threadIdx.x] += red[512 + threadIdx.x + off];
    }
    __syncthreads();
  }
  if (threadIdx.x == 0) {
    const double s = 1.0 / ((double)N * (double)(N - 3));
    const double gxy = red[0] * s, gxx = red[256] * s, gyy = red[512] * s;
    const double corr = gxy / sqrt(gxx * gyy + 1e-18);
    out[0] = (float)(-corr);
  }
}

extern "C" void kernel_launch(void* const* d_in, const int* in_sizes, int n_in,
                              void* d_out, int out_size, void* d_ws, size_t ws_size,
                              hipStream_t stream) {
  const float* X = (const float*)d_in[0];
  const float* Y = (const float*)d_in[1];
  float* out = (float*)d_out;

  const size_t nn = (size_t)N * N;
  float* A     = (float*)d_ws;            // n*n distance matrix for X
  float* B     = A + nn;                  // n*n distance matrix for Y
  float* sqX   = B + nn;                  // N
  float* sqY   = sqX + N;                 // N
  float* rA    = sqY + N;                 // N  (column sums of A)
  float* rB    = rA + N;                  // N
  float* partA = rB + N;                  // ICH*N
  float* partB = partA + (size_t)ICH * N; // ICH*N
  double* tots = (double*)(partB + (size_t)ICH * N); // 2 doubles (8B-aligned: even float count)
  double* bpart = tots + 2;               // BBLOCKS*3 doubles

  rowsq_kernel<<<N, 256, 0, stream>>>(X, sqX);
  rowsq_kernel<<<N, 256, 0, stream>>>(Y, sqY);

  // 4096 64x64 tiles, 8 waves/block -> 512 blocks
  gram_dist_kernel<<<(N / 64) * (N / 64) / 8, 256, 0, stream>>>(X, sqX, A);
  gram_dist_kernel<<<(N / 64) * (N / 64) / 8, 256, 0, stream>>>(Y, sqY, B);

  dim3 gcs(N / 256, ICH);
  colsum_part_kernel<<<gcs, 256, 0, stream>>>(A, partA);
  colsum_part_kernel<<<gcs, 256, 0, stream>>>(B, partB);
  colsum_final_kernel<<<N / 256, 256, 0, stream>>>(partA, rA);
  colsum_final_kernel<<<N / 256, 256, 0, stream>>>(partB, rB);
  tot_kernel<<<1, 256, 0, stream>>>(rA, rB, tots);

  bracket_kernel<<<BBLOCKS, 256, 0, stream>>>(A, B, rA, rB, tots, bpart);
  final_kernel<<<1, 256, 0, stream>>>(bpart, out);
}